// Conv4dNet_40183714021983
// MI455X (gfx1250) — compile-verified
//
#include <hip/hip_runtime.h>

typedef __attribute__((ext_vector_type(2))) float v2f;
typedef __attribute__((ext_vector_type(4))) float v4f;
typedef __attribute__((ext_vector_type(8))) float v8f;

#define P_TOT 38416   // 14^4 real positions
#define P_PAD 38528   // 16*2408, divisible by 128 (padded position stride)
#define S1 14
#define S2 196        // 14^2
#define S3 2744       // 14^3
// activations stored interleaved: [C/4][P_PAD][4]
#define ACT_ELEMS (160 * P_PAD)
#define WBUF_ELEMS (81 * 160 * 96)

static __device__ __forceinline__ v8f wmma4(v2f a, v2f b, v8f c) {
  return __builtin_amdgcn_wmma_f32_16x16x4_f32(
      /*neg_a=*/false, a, /*neg_b=*/false, b,
      /*c_mod=*/(short)0, c, /*reuse_a=*/false, /*reuse_b=*/false);
}

// ---- input prep: (group0, c4==0, p<P_TOT) <- x, everything else 0 ----------
// addr == idx for the [cg][p][4] layout, so this also zeroes all tails.
__global__ void prep_input(const float* __restrict__ x, float* __restrict__ act) {
  int idx = blockIdx.x * blockDim.x + threadIdx.x;
  if (idx >= 16 * P_PAD) return;
  int cg = idx / (4 * P_PAD);
  int r  = idx % (4 * P_PAD);
  int p  = r >> 2, c4 = r & 3;
  act[idx] = (cg == 0 && c4 == 0 && p < P_TOT) ? x[p] : 0.0f;
}

// ---- weight repack: [Co][Ci][81] -> [81][Cop][Cip] (zero pad) --------------
__global__ void pack_w(const float* __restrict__ w, float* __restrict__ wp,
                       int Ci, int Co, int Cip, int Cop) {
  int idx = blockIdx.x * blockDim.x + threadIdx.x;
  int tot = 81 * Cop * Cip;
  if (idx >= tot) return;
  int ci = idx % Cip;
  int co = (idx / Cip) % Cop;
  int t  = idx / (Cip * Cop);
  wp[idx] = (co < Co && ci < Ci) ? w[(co * Ci + ci) * 81 + t] : 0.0f;
}

// ---- implicit-GEMM conv4d via V_WMMA_F32_16X16X4_F32 -----------------------
// Block = 4 waves sharing co range [m0,m0+32); per-tap A panel staged in LDS.
// Each wave: 32x32 register tile (4 v8f accumulators). Activations are in
// C4-interleaved layout so each B fragment is ONE b64 load; border zeros come
// from the always-zero tail positions [P_TOT,P_PAD) -> no divergent loads.
__global__ void __launch_bounds__(128) conv4d_wmma(
    const float* __restrict__ act_in, const float* __restrict__ wp,
    float* __restrict__ act_out, int Cip, int Cop)
{
  __shared__ float ldsA[32 * 160];    // 20 KB max (Cip <= 160)
  const int tid  = threadIdx.x;
  const int wv   = tid >> 5;
  const int lane = tid & 31;
  const int half = lane >> 4;
  const int l    = lane & 15;
  const int m0   = blockIdx.y * 32;
  const int pt0  = blockIdx.x * 128 + wv * 32;

  int pn[2];  pn[0] = pt0 + l;  pn[1] = pt0 + 16 + l;
  int ciX[2], cjX[2], ckX[2], clX[2]; bool inb[2];
#pragma unroll
  for (int n = 0; n < 2; ++n) {
    inb[n] = pn[n] < P_TOT;
    int pp = inb[n] ? pn[n] : 0;
    ciX[n] = pp / S3; int r = pp % S3;
    cjX[n] = r / S2;  r %= S2;
    ckX[n] = r / S1;  clX[n] = r % S1;
  }

  v8f acc00 = {}, acc01 = {}, acc10 = {}, acc11 = {};
  const int kg = Cip >> 2;                      // K groups of 4 channels

  for (int ta = 0; ta < 3; ++ta)
  for (int tb = 0; tb < 3; ++tb)
  for (int tc = 0; tc < 3; ++tc)
  for (int td = 0; td < 3; ++td) {
    const int t = ((ta * 3 + tb) * 3 + tc) * 3 + td;

    // stage A panel wp[t][m0:m0+32][0:Cip] -> LDS (coalesced, 4 waves share)
    const float* wt = wp + (size_t)t * Cop * Cip + (size_t)m0 * Cip;
    __syncthreads();
    for (int idx = tid; idx < 32 * Cip; idx += 128) ldsA[idx] = wt[idx];
    __syncthreads();

    // shifted source positions; invalid -> P_TOT (always-zero tail)
    int q[2];
#pragma unroll
    for (int n = 0; n < 2; ++n) {
      int qi = ciX[n] + ta - 1, qj = cjX[n] + tb - 1;
      int qk = ckX[n] + tc - 1, ql = clX[n] + td - 1;
      bool valid = inb[n] && ((unsigned)qi < 14u) && ((unsigned)qj < 14u) &&
                   ((unsigned)qk < 14u) && ((unsigned)ql < 14u);
      q[n] = valid ? ((qi * 14 + qj) * 14 + qk) * 14 + ql : P_TOT;
    }

    const float* pB0 = act_in + (size_t)q[0] * 4 + 2 * half;
    const float* pB1 = act_in + (size_t)q[1] * 4 + 2 * half;
    const float* pA  = ldsA + 4 * 0 + 2 * half;

    for (int k = 0; k < kg; ++k) {
      v2f a0 = *(const v2f*)(ldsA + l * Cip        + 4 * k + 2 * half);
      v2f a1 = *(const v2f*)(ldsA + (16 + l) * Cip + 4 * k + 2 * half);
      v2f b0 = *(const v2f*)pB0;
      v2f b1 = *(const v2f*)pB1;
      pB0 += 4 * (size_t)P_PAD;
      pB1 += 4 * (size_t)P_PAD;
      acc00 = wmma4(a0, b0, acc00);
      acc01 = wmma4(a0, b1, acc01);
      acc10 = wmma4(a1, b0, acc10);
      acc11 = wmma4(a1, b1, acc11);
    }
    (void)pA;
  }

  // write out: rows m0+8*half+{0..3} are one contiguous 16B chunk per position
  const int gbase = (m0 + 8 * half) >> 2;        // channel group of acc*[0..3]
  float* o0 = act_out + (size_t)gbase * 4 * P_PAD;
  v4f lo00 = {acc00[0], acc00[1], acc00[2], acc00[3]};
  v4f hi00 = {acc00[4], acc00[5], acc00[6], acc00[7]};
  v4f lo01 = {acc01[0], acc01[1], acc01[2], acc01[3]};
  v4f hi01 = {acc01[4], acc01[5], acc01[6], acc01[7]};
  v4f lo10 = {acc10[0], acc10[1], acc10[2], acc10[3]};
  v4f hi10 = {acc10[4], acc10[5], acc10[6], acc10[7]};
  v4f lo11 = {acc11[0], acc11[1], acc11[2], acc11[3]};
  v4f hi11 = {acc11[4], acc11[5], acc11[6], acc11[7]};
  *(v4f*)(o0 + (size_t)pn[0] * 4)                 = lo00;   // group gbase
  *(v4f*)(o0 + 4 * (size_t)P_PAD + pn[0] * 4)     = hi00;   // group gbase+1
  *(v4f*)(o0 + (size_t)pn[1] * 4)                 = lo01;
  *(v4f*)(o0 + 4 * (size_t)P_PAD + pn[1] * 4)     = hi01;
  float* o1 = o0 + 16 * (size_t)P_PAD;                      // +16 rows = +4 groups
  *(v4f*)(o1 + (size_t)pn[0] * 4)                 = lo10;
  *(v4f*)(o1 + 4 * (size_t)P_PAD + pn[0] * 4)     = hi10;
  *(v4f*)(o1 + (size_t)pn[1] * 4)                 = lo11;
  *(v4f*)(o1 + 4 * (size_t)P_PAD + pn[1] * 4)     = hi11;
}

// ---- BN statistics: one block per channel GROUP (4 channels) ---------------
__global__ void bn_stats(const float* __restrict__ act, float* __restrict__ stats,
                         int Creal) {
  const int grp = blockIdx.x;
  const int c4  = threadIdx.x & 3;
  const int pi  = threadIdx.x >> 2;              // 0..63
  __shared__ float ss[256], sq[256];
  float s = 0.f, q = 0.f;
  const float* base = act + (size_t)grp * 4 * P_PAD + c4;
  for (int p = pi; p < P_TOT; p += 64) {
    float v = base[(size_t)p * 4];
    s += v; q += v * v;
  }
  const int me = c4 * 64 + pi;
  ss[me] = s; sq[me] = q;
  __syncthreads();
  for (int o = 32; o > 0; o >>= 1) {
    if (pi < o) { ss[me] += ss[me + o]; sq[me] += sq[me + o]; }
    __syncthreads();
  }
  if (pi == 0) {
    int c = grp * 4 + c4;
    if (c < Creal) {
      float mean = ss[me] / (float)P_TOT;
      float var  = sq[me] / (float)P_TOT - mean * mean;
      stats[2 * c]     = mean;
      stats[2 * c + 1] = rsqrtf(var + 1e-5f);
    }
  }
}

// ---- BN affine + ReLU (in place), zero padded channels, coalesced ----------
__global__ void bn_relu(float* __restrict__ act, const float* __restrict__ stats,
                        const float* __restrict__ g, const float* __restrict__ bt,
                        int Cp, int Creal) {
  int idx = blockIdx.x * blockDim.x + threadIdx.x;
  if (idx >= Cp * P_TOT) return;
  int cg = idx / (4 * P_TOT);
  int r  = idx % (4 * P_TOT);
  int p  = r >> 2, c4 = r & 3;
  int c  = cg * 4 + c4;
  size_t a = (size_t)cg * 4 * P_PAD + (size_t)p * 4 + c4;
  float v = 0.f;
  if (c < Creal) {
    float x = act[a];
    v = (x - stats[2 * c]) * stats[2 * c + 1] * g[c] + bt[c];
    v = v > 0.f ? v : 0.f;
  }
  act[a] = v;
}

// ---- final: bias + ReLU on channel 0 (group 0, c4 0) -----------------------
__global__ void bias_relu_out(const float* __restrict__ act,
                              const float* __restrict__ b6,
                              float* __restrict__ out) {
  int p = blockIdx.x * blockDim.x + threadIdx.x;
  if (p >= P_TOT) return;
  float v = act[(size_t)p * 4] + b6[0];
  out[p] = v > 0.f ? v : 0.f;
}

extern "C" void kernel_launch(void* const* d_in, const int* in_sizes, int n_in,
                              void* d_out, int out_size, void* d_ws, size_t ws_size,
                              hipStream_t stream) {
  (void)in_sizes; (void)n_in; (void)out_size; (void)ws_size;
  const float* x  = (const float*)d_in[0];
  const float* w[6]  = {(const float*)d_in[1], (const float*)d_in[2], (const float*)d_in[3],
                        (const float*)d_in[4], (const float*)d_in[5], (const float*)d_in[6]};
  const float* g[5]  = {(const float*)d_in[7],  (const float*)d_in[9],  (const float*)d_in[11],
                        (const float*)d_in[13], (const float*)d_in[15]};
  const float* bb[5] = {(const float*)d_in[8],  (const float*)d_in[10], (const float*)d_in[12],
                        (const float*)d_in[14], (const float*)d_in[16]};
  const float* b6 = (const float*)d_in[17];
  float* out = (float*)d_out;

  float* ws    = (float*)d_ws;
  float* actA  = ws;
  float* actB  = ws + ACT_ELEMS;
  float* wbuf  = ws + 2 * (size_t)ACT_ELEMS;
  float* stats = wbuf + WBUF_ELEMS;

  static const int Ci[6]  = {1, 40, 80, 160, 80, 40};
  static const int Co[6]  = {40, 80, 160, 80, 40, 1};
  static const int Cip[6] = {16, 64, 96, 160, 96, 64};
  static const int Cop[6] = {64, 96, 160, 96, 64, 32};

  prep_input<<<(16 * P_PAD + 255) / 256, 256, 0, stream>>>(x, actA);

  float* cur = actA;
  float* nxt = actB;
  for (int L = 0; L < 6; ++L) {
    int tot = 81 * Cop[L] * Cip[L];
    pack_w<<<(tot + 255) / 256, 256, 0, stream>>>(w[L], wbuf, Ci[L], Co[L], Cip[L], Cop[L]);

    dim3 grid(P_PAD / 128, Cop[L] / 32);
    conv4d_wmma<<<grid, 128, 0, stream>>>(cur, wbuf, nxt, Cip[L], Cop[L]);

    if (L < 5) {
      bn_stats<<<(Co[L] + 3) / 4, 256, 0, stream>>>(nxt, stats, Co[L]);
      int n = Cop[L] * P_TOT;
      bn_relu<<<(n + 255) / 256, 256, 0, stream>>>(nxt, stats, g[L], bb[L], Cop[L], Co[L]);
    } else {
      bias_relu_out<<<(P_TOT + 255) / 256, 256, 0, stream>>>(nxt, b6, out);
    }
    float* t = cur; cur = nxt; nxt = t;
  }
}